// DCTLayer_2052994367489
// MI455X (gfx1250) — compile-verified
//
#include <hip/hip_runtime.h>
#include <math.h>

typedef float v2f __attribute__((ext_vector_type(2)));
typedef float v8f __attribute__((ext_vector_type(8)));

#define LDS_STRIDE 20      // floats: float4 slots stay 16B aligned; column reads are bank-conflict-free
#define WAVES_PER_BLOCK 8

__device__ __forceinline__ void wave_lds_fence() {
    // LDS is wave-private per tile buffer; just need DS completion + compiler ordering.
    asm volatile("s_wait_dscnt 0" ::: "memory");
}

__global__ __launch_bounds__(256)
void dct8x8_wmma_kernel(const float* __restrict__ x, float* __restrict__ out,
                        int imgW, int n_tiles_x, int tiles_per_img, long long total_tiles) {
    __shared__ float T_lds[16 * LDS_STRIDE];
    __shared__ float x_tile[WAVES_PER_BLOCK][16 * LDS_STRIDE];
    __shared__ float y_tile[WAVES_PER_BLOCK][16 * LDS_STRIDE];

    const int tid = threadIdx.x;

    // Build block-diagonal orthonormal DCT-II transform T = diag(D8, D8), 16x16.
    // D[k][m] = 0.5*cos(pi*(2m+1)*k/16) for k>0, D[0][m] = 0.5*sqrt(0.5).
    {
        const int i = tid >> 4, j = tid & 15;   // 256 threads cover 16x16
        float v = 0.0f;
        if ((i < 8) == (j < 8)) {
            const int k = i & 7, mm = j & 7;
            v = (k == 0) ? 0.35355339059327373f
                         : 0.5f * cosf(3.14159265358979323846f * (2.0f * mm + 1.0f) * (float)k * 0.0625f);
        }
        T_lds[i * LDS_STRIDE + j] = v;
    }
    __syncthreads();

    const int wave = tid >> 5;
    const int lane = tid & 31;
    const bool lo  = lane < 16;
    const int  m   = lane & 15;

    const long long gwave = (long long)blockIdx.x * WAVES_PER_BLOCK + wave;
    if (gwave >= total_tiles) return;   // wave-uniform: EXEC stays all-ones for active waves

    // T fragments: A-frag of T chunk k == B-frag of T^T chunk k, so load once, use twice.
    v2f tf[4];
#pragma unroll
    for (int k = 0; k < 4; ++k) {
        const int kb = 4 * k + (lo ? 0 : 2);
        tf[k].x = T_lds[m * LDS_STRIDE + kb];
        tf[k].y = T_lds[m * LDS_STRIDE + kb + 1];
    }

    const int img = (int)(gwave / tiles_per_img);
    const int t   = (int)(gwave % tiles_per_img);
    const int th  = t / n_tiles_x, tw = t % n_tiles_x;
    const long long base = (long long)img * imgW * imgW + (long long)th * 16 * imgW + (long long)tw * 16;

    float* xl = x_tile[wave];
    float* yl = y_tile[wave];

    // ---- Load 16x16 fp32 tile: two b128 loads per lane, coalesced ----
#pragma unroll
    for (int i = 0; i < 2; ++i) {
        const int q = lane + 32 * i;              // 0..63 float4 slots
        const int row = q >> 2, c4 = (q & 3) << 2;
        const float4 v = *(const float4*)(x + base + (long long)row * imgW + c4);
        *(float4*)(xl + row * LDS_STRIDE + c4) = v;
    }
    wave_lds_fence();

    // ---- GEMM1: Y = T @ X  (4 chained v_wmma_f32_16x16x4_f32, fp32 accumulate) ----
    v8f acc = {};
#pragma unroll
    for (int k = 0; k < 4; ++k) {
        const int kb = 4 * k + (lo ? 0 : 2);
        v2f b;
        b.x = xl[kb * LDS_STRIDE + m];
        b.y = xl[(kb + 1) * LDS_STRIDE + m];
        acc = __builtin_amdgcn_wmma_f32_16x16x4_f32(false, tf[k], false, b,
                                                    (short)0, acc, false, false);
    }

    // Spill Y (C layout: vgpr r -> row r / r+8) so it can be re-read in A layout.
#pragma unroll
    for (int r = 0; r < 8; ++r) {
        const int row = r + (lo ? 0 : 8);
        yl[row * LDS_STRIDE + m] = acc[r];
    }
    wave_lds_fence();

    // ---- GEMM2: OUT = Y @ T^T  (B-frag of T^T is tf[k]) ----
    v8f acc2 = {};
#pragma unroll
    for (int k = 0; k < 4; ++k) {
        const int kb = 4 * k + (lo ? 0 : 2);
        v2f a;
        a.x = yl[m * LDS_STRIDE + kb];
        a.y = yl[m * LDS_STRIDE + kb + 1];
        acc2 = __builtin_amdgcn_wmma_f32_16x16x4_f32(false, a, false, tf[k],
                                                     (short)0, acc2, false, false);
    }

    // ---- Stage OUT in LDS (C layout), then coalesced b128 stores ----
#pragma unroll
    for (int r = 0; r < 8; ++r) {
        const int row = r + (lo ? 0 : 8);
        xl[row * LDS_STRIDE + m] = acc2[r];
    }
    wave_lds_fence();

#pragma unroll
    for (int i = 0; i < 2; ++i) {
        const int q = lane + 32 * i;
        const int row = q >> 2, c4 = (q & 3) << 2;
        const float4 v = *(const float4*)(xl + row * LDS_STRIDE + c4);
        *(float4*)(out + base + (long long)row * imgW + c4) = v;
    }
}

extern "C" void kernel_launch(void* const* d_in, const int* in_sizes, int n_in,
                              void* d_out, int out_size, void* d_ws, size_t ws_size,
                              hipStream_t stream) {
    const float* x = (const float*)d_in[0];
    float* out = (float*)d_out;

    const int imgW = 512;                                 // H == W == 512
    const int n_imgs = in_sizes[0] / (imgW * imgW);       // B*C = 96
    const int ntx = imgW / 16;                            // 32
    const int tiles_per_img = ntx * ntx;                  // 1024
    const long long total_tiles = (long long)n_imgs * tiles_per_img;  // 98304

    const long long waves = total_tiles;
    const int blocks = (int)((waves + WAVES_PER_BLOCK - 1) / WAVES_PER_BLOCK);  // 12288

    dct8x8_wmma_kernel<<<blocks, 256, 0, stream>>>(x, out, imgW, ntx, tiles_per_img, total_tiles);
}